// PointnetSAModuleMSG_52896817217689
// MI455X (gfx1250) — compile-verified
//
#include <hip/hip_runtime.h>
#include <hip/hip_bf16.h>

typedef __attribute__((ext_vector_type(2))) float v2f;
typedef __attribute__((ext_vector_type(8))) float v8f;

#define NB      8       // batches
#define NPTS    4096    // points per batch
#define NQ      1024    // npoint (queries)
#define CFEAT   64      // feature channels
#define CTOT    640     // 128 + 256 + 256 output channels
#define STRIDE  132     // LDS row stride (floats), even => 8B-aligned v2f rows

struct LayerP { const float *w, *bias, *gamma, *beta, *mean, *var; };

// ---------------------------------------------------------------------------
// 1) Furthest point sampling: one block per batch. dist/xyz live in LDS.
//    Matches jax scan semantics: emit `last`, update dist with xyz[last],
//    next = argmax(dist) (first occurrence on ties).
// ---------------------------------------------------------------------------
__global__ __launch_bounds__(1024) void fps_kernel(const float* __restrict__ xyz,
                                                   float* __restrict__ newxyz) {
    __shared__ float sx[NPTS], sy[NPTS], sz[NPTS], dist[NPTS];
    __shared__ float rv[32];
    __shared__ int   ri[32];
    const int b = blockIdx.x, tid = threadIdx.x;
    for (int j = tid; j < NPTS; j += 1024) {
        sx[j] = xyz[(b * NPTS + j) * 3 + 0];
        sy[j] = xyz[(b * NPTS + j) * 3 + 1];
        sz[j] = xyz[(b * NPTS + j) * 3 + 2];
        dist[j] = 1e10f;
    }
    __syncthreads();
    int last = 0;
    for (int it = 0; it < NQ; ++it) {
        if (tid == 0) {
            newxyz[(b * NQ + it) * 3 + 0] = sx[last];
            newxyz[(b * NQ + it) * 3 + 1] = sy[last];
            newxyz[(b * NQ + it) * 3 + 2] = sz[last];
        }
        const float lx = sx[last], ly = sy[last], lz = sz[last];
        float bv = -1.0f; int bi_ = 0;
        for (int j = tid; j < NPTS; j += 1024) {
            float dx = sx[j] - lx, dy = sy[j] - ly, dz = sz[j] - lz;
            float nd = fminf(dist[j], dx * dx + dy * dy + dz * dz);
            dist[j] = nd;
            if (nd > bv) { bv = nd; bi_ = j; }   // strict > keeps smallest j
        }
        // wave32 reduction (max value, smallest index on tie)
        for (int off = 16; off > 0; off >>= 1) {
            float ov = __shfl_xor(bv, off, 32);
            int   oi = __shfl_xor(bi_, off, 32);
            if (ov > bv || (ov == bv && oi < bi_)) { bv = ov; bi_ = oi; }
        }
        if ((tid & 31) == 0) { rv[tid >> 5] = bv; ri[tid >> 5] = bi_; }
        __syncthreads();
        if (tid < 32) {
            bv = rv[tid]; bi_ = ri[tid];
            for (int off = 16; off > 0; off >>= 1) {
                float ov = __shfl_xor(bv, off, 32);
                int   oi = __shfl_xor(bi_, off, 32);
                if (ov > bv || (ov == bv && oi < bi_)) { bv = ov; bi_ = oi; }
            }
            if (tid == 0) ri[0] = bi_;
        }
        __syncthreads();
        last = ri[0];
        __syncthreads();
    }
}

// ---------------------------------------------------------------------------
// 2) Ball query: one thread per query, xyz tiled through LDS.
//    First nsample hits in ascending index order; pad with first hit;
//    zero hits -> all N-1 (matches reference clip).
// ---------------------------------------------------------------------------
__global__ __launch_bounds__(128) void ballq_kernel(const float* __restrict__ xyz,
                                                    const float* __restrict__ newxyz,
                                                    int* __restrict__ idxbuf,
                                                    float r2, int ns) {
    __shared__ float sx[1024], sy[1024], sz[1024];
    const int b = blockIdx.x / (NQ / 128);
    const int s = (blockIdx.x % (NQ / 128)) * 128 + threadIdx.x;
    const float qx = newxyz[(b * NQ + s) * 3 + 0];
    const float qy = newxyz[(b * NQ + s) * 3 + 1];
    const float qz = newxyz[(b * NQ + s) * 3 + 2];
    int* myidx = idxbuf + (b * NQ + s) * ns;
    int cnt = 0;
    for (int tile = 0; tile < NPTS / 1024; ++tile) {
        __syncthreads();
        for (int t = threadIdx.x; t < 1024; t += 128) {
            int j = tile * 1024 + t;
            sx[t] = xyz[(b * NPTS + j) * 3 + 0];
            sy[t] = xyz[(b * NPTS + j) * 3 + 1];
            sz[t] = xyz[(b * NPTS + j) * 3 + 2];
        }
        __syncthreads();
        if (cnt < ns) {
            for (int t = 0; t < 1024 && cnt < ns; ++t) {
                float dx = sx[t] - qx, dy = sy[t] - qy, dz = sz[t] - qz;
                if (dx * dx + dy * dy + dz * dz < r2) myidx[cnt++] = tile * 1024 + t;
            }
        }
    }
    const int fill = (cnt == 0) ? (NPTS - 1) : myidx[0];
    for (int k = cnt; k < ns; ++k) myidx[k] = fill;
}

// ---------------------------------------------------------------------------
// 3) Fused grouping + 3-layer MLP (+BN+ReLU) + neighbor max-pool.
//    GEMM via V_WMMA_F32_16X16X4_F32. A (rows x K) staged in LDS; B (=W^T)
//    fragments preloaded into VGPRs once per col-tile and reused across all
//    row tiles / query groups.
//    Fragment layout per ISA 7.12.2:
//      A 16x4 f32: lanes 0-15 -> K={kb,kb+1}, lanes 16-31 -> K={kb+2,kb+3}
//      B  4x16:    lanes 0-15 -> K={kb,kb+1}, lanes 16-31 -> K={kb+2,kb+3},
//                  col = lane%16
//      C/D 16x16:  VGPR v: row = v + 8*(lane/16), col = lane%16
// ---------------------------------------------------------------------------
template <int CIN, int KPAD>
__device__ inline void load_bfrags(const float* __restrict__ W, int ct, int lane,
                                   v2f* __restrict__ bf) {
    const int koff = (lane >> 4) << 1;
    const float* wr = W + (ct * 16 + (lane & 15)) * CIN;
#pragma unroll
    for (int ks = 0; ks < KPAD / 4; ++ks) {
        const int k0 = ks * 4 + koff;
        if constexpr (CIN % 4 == 0) {
            bf[ks] = *(const v2f*)(wr + k0);          // 8B-aligned: CIN%4==0, k0 even
        } else {
            v2f bb;
            bb.x = (k0 < CIN)     ? wr[k0]     : 0.f; // zero-pad K tail
            bb.y = (k0 + 1 < CIN) ? wr[k0 + 1] : 0.f;
            bf[ks] = bb;
        }
    }
}

template <int KPAD>
__device__ inline v8f wmma_kloop(const float (*bi)[STRIDE], const v2f* __restrict__ bf,
                                 int rt, int lane) {
    v8f acc = {0.f, 0.f, 0.f, 0.f, 0.f, 0.f, 0.f, 0.f};
    const int koff = (lane >> 4) << 1;
    const int arow = rt * 16 + (lane & 15);
#pragma unroll
    for (int ks = 0; ks < KPAD / 4; ++ks) {
        const v2f a = *(const v2f*)(&bi[arow][ks * 4 + koff]);   // ds_load_b64
        acc = __builtin_amdgcn_wmma_f32_16x16x4_f32(false, a, false, bf[ks],
                                                    (short)0, acc, false, false);
    }
    return acc;
}

template <int CIN, int KPAD, int COUT>
__device__ inline void mlp_layer_mid(const LayerP& P, const float (*bi)[STRIDE],
                                     float (*bo)[STRIDE], int lane, int wave) {
    const int NT = COUT / 16;
    v2f bf[KPAD / 4];
    for (int ct = wave; ct < NT; ct += 4) {
        load_bfrags<CIN, KPAD>(P.w, ct, lane, bf);
        const int c  = ct * 16 + (lane & 15);
        const float sc = P.gamma[c] / sqrtf(P.var[c] + 1e-5f);
        const float sh = P.beta[c] + (P.bias[c] - P.mean[c]) * sc;
        for (int rt = 0; rt < 4; ++rt) {
            v8f acc = wmma_kloop<KPAD>(bi, bf, rt, lane);
            const int rbase = rt * 16 + ((lane >> 4) << 3);
#pragma unroll
            for (int v = 0; v < 8; ++v)
                bo[rbase + v][c] = fmaxf(acc[v] * sc + sh, 0.f);
        }
    }
}

template <int CIN, int KPAD, int COUT, int NS>
__device__ inline void mlp_layer_last(const LayerP& P, const float (*bi)[STRIDE],
                                      float* __restrict__ outFeat, int b, int sbase,
                                      int cbase, int lane, int wave) {
    const int NT = COUT / 16;
    const int G  = NS / 16;    // row tiles per query group
    const int QW = 64 / NS;    // query groups per workgroup
    v2f bf[KPAD / 4];
    for (int ct = wave; ct < NT; ct += 4) {
        load_bfrags<CIN, KPAD>(P.w, ct, lane, bf);
        const int c  = ct * 16 + (lane & 15);
        const float sc = P.gamma[c] / sqrtf(P.var[c] + 1e-5f);
        const float sh = P.beta[c] + (P.bias[c] - P.mean[c]) * sc;
        for (int q = 0; q < QW; ++q) {
            float m = 0.f;   // relu outputs are >= 0
            for (int g = 0; g < G; ++g) {
                v8f acc = wmma_kloop<KPAD>(bi, bf, q * G + g, lane);
#pragma unroll
                for (int v = 0; v < 8; ++v)
                    m = fmaxf(m, fmaxf(acc[v] * sc + sh, 0.f));
            }
            m = fmaxf(m, __shfl_xor(m, 16, 32));   // combine rows 0-7 with 8-15
            if (lane < 16)
                outFeat[((size_t)b * CTOT + (cbase + ct * 16 + lane)) * NQ + (sbase + q)] = m;
        }
    }
}

template <int NS, int C1, int C2, int C3, int CBASE>
__global__ __launch_bounds__(128) void msg_mlp_kernel(const float* __restrict__ xyz,
                                                      const float* __restrict__ feats,
                                                      const float* __restrict__ newxyz,
                                                      const int* __restrict__ idxbuf,
                                                      LayerP L1, LayerP L2, LayerP L3,
                                                      float* __restrict__ outFeat) {
    __shared__ float bufA[64][STRIDE];
    __shared__ float bufB[64][STRIDE];
    const int QW    = 64 / NS;                         // queries per workgroup
    const int b     = blockIdx.x / (NQ / QW);
    const int sbase = (blockIdx.x % (NQ / QW)) * QW;
    const int tid   = threadIdx.x;
    const int lane  = tid & 31, wave = tid >> 5;

    // Stage grouped input: [g_xyz(3) | g_feat(64) | zero-pad(1)] per row.
    for (int e = tid; e < 64 * 68; e += 128) {
        const int r = e / 68, c = e % 68;
        const int q = r / NS, n = r % NS;
        const int s = sbase + q;
        const int j = idxbuf[(b * NQ + s) * NS + n];
        float val;
        if (c < 3)       val = xyz[(b * NPTS + j) * 3 + c] - newxyz[(b * NQ + s) * 3 + c];
        else if (c < 67) val = feats[((size_t)b * CFEAT + (c - 3)) * NPTS + j];
        else             val = 0.f;
        bufA[r][c] = val;
    }
    __syncthreads();
    mlp_layer_mid<67, 68, C1>(L1, bufA, bufB, lane, wave);
    __syncthreads();
    mlp_layer_mid<C1, C1, C2>(L2, bufB, bufA, lane, wave);
    __syncthreads();
    mlp_layer_last<C2, C2, C3, NS>(L3, bufA, outFeat, b, sbase, CBASE, lane, wave);
}

// ---------------------------------------------------------------------------
// Launch. d_in order (setup_inputs dict order): xyz, features, then
// params[scale][layer] leaves in insertion order: w, b, gamma, beta, mean, var.
// d_out = new_xyz[8,1024,3] ++ new_features[8,640,1024] (flat).
// ---------------------------------------------------------------------------
extern "C" void kernel_launch(void* const* d_in, const int* in_sizes, int n_in,
                              void* d_out, int out_size, void* d_ws, size_t ws_size,
                              hipStream_t stream) {
    (void)in_sizes; (void)n_in; (void)out_size; (void)ws_size;
    const float* xyz   = (const float*)d_in[0];
    const float* feats = (const float*)d_in[1];
    auto getL = [&](int scale, int l) {
        const int base = 2 + (scale * 3 + l) * 6;
        LayerP p;
        p.w     = (const float*)d_in[base + 0];
        p.bias  = (const float*)d_in[base + 1];
        p.gamma = (const float*)d_in[base + 2];
        p.beta  = (const float*)d_in[base + 3];
        p.mean  = (const float*)d_in[base + 4];
        p.var   = (const float*)d_in[base + 5];
        return p;
    };
    float* out     = (float*)d_out;
    float* newxyz  = out;                       // 8*1024*3
    float* outFeat = out + NB * NQ * 3;         // 8*640*1024
    int* idx1 = (int*)d_ws;                     // 8*1024*16
    int* idx2 = idx1 + NB * NQ * 16;            // 8*1024*32
    int* idx3 = idx2 + NB * NQ * 32;            // 8*1024*64

    fps_kernel<<<NB, 1024, 0, stream>>>(xyz, newxyz);

    ballq_kernel<<<NB * (NQ / 128), 128, 0, stream>>>(xyz, newxyz, idx1, 0.1f * 0.1f, 16);
    ballq_kernel<<<NB * (NQ / 128), 128, 0, stream>>>(xyz, newxyz, idx2, 0.2f * 0.2f, 32);
    ballq_kernel<<<NB * (NQ / 128), 128, 0, stream>>>(xyz, newxyz, idx3, 0.4f * 0.4f, 64);

    msg_mlp_kernel<16, 64, 64, 128, 0><<<NB * NQ / 4, 128, 0, stream>>>(
        xyz, feats, newxyz, idx1, getL(0, 0), getL(0, 1), getL(0, 2), outFeat);
    msg_mlp_kernel<32, 128, 128, 256, 128><<<NB * NQ / 2, 128, 0, stream>>>(
        xyz, feats, newxyz, idx2, getL(1, 0), getL(1, 1), getL(1, 2), outFeat);
    msg_mlp_kernel<64, 128, 128, 256, 384><<<NB * NQ, 128, 0, stream>>>(
        xyz, feats, newxyz, idx3, getL(2, 0), getL(2, 1), getL(2, 2), outFeat);
}